// SimplePCNet_41386304864898
// MI455X (gfx1250) — compile-verified
//
#include <hip/hip_runtime.h>

typedef __attribute__((ext_vector_type(2))) float v2f;
typedef __attribute__((ext_vector_type(8))) float v8f;

// One wave32 computes a 16-row output tile of  Y[j,:] = sum_k mask * X[G[k,j],:] @ W[k]
// with V_WMMA_F32_16X16X4_F32 (full fp32 precision, K=4 per step).
//
// Operand layouts (CDNA5 ISA 7.12.2, wave32):
//   A 16x4  f32: lane L (m=L&15, hi=L>>4) holds {A[m,2hi], A[m,2hi+1]}        (2 VGPRs)
//   B 4x16  f32: lane L (n=L&15, hi=L>>4) holds {B[2hi,n], B[2hi+1,n]}        (2 VGPRs)
//   D 16x16 f32: VGPR r, lane L (n=L&15, hi)  = D[r + 8*hi, n]                (8 VGPRs)
//
// W is staged once per block into LDS, permuted into per-lane operand order so
// each B fetch is a single ds_load_b64 with an immediate offset:
//   wlds[((k*KSTEPS + kk)*2 + hi)*COUT + n] = { W[k][4kk+2hi][n], W[k][4kk+2hi+1][n] }
template <int CIN, int COUT>
__global__ void __launch_bounds__(256) sparse_conv_wmma(
    const float* __restrict__ X,  // [*, CIN] gather source
    const float* __restrict__ W,  // [27, CIN, COUT]
    const int*   __restrict__ G,  // [27, M] gather rows (0 on miss)
    const int*   __restrict__ S,  // [27, M] == j when hit, == M (sentinel) on miss
    float*       __restrict__ Y,  // [M, COUT]
    int M) {
  constexpr int KSTEPS = CIN / 4;
  constexpr int NCH    = (COUT + 15) / 16;   // 16-wide column chunks
  constexpr int PAIRS  = 27 * KSTEPS * 2 * COUT;

  extern __shared__ v2f wlds[];

  // ---- cooperative W preload into permuted LDS image (one-time, coalesced-ish)
  for (int p = (int)threadIdx.x; p < PAIRS; p += (int)blockDim.x) {
    const int n  = p % COUT;
    const int t  = p / COUT;
    const int h  = t & 1;
    const int kk = (t >> 1) % KSTEPS;
    const int k  = (t >> 1) / KSTEPS;
    const float* src = W + (long)(k * CIN + kk * 4 + 2 * h) * COUT + n;
    v2f w;
    w.x = src[0];
    w.y = src[COUT];
    wlds[p] = w;
  }
  __syncthreads();  // all waves reach this before any tail-wave exit

  const int wave = (int)((blockIdx.x * blockDim.x + threadIdx.x) >> 5);
  const int lane = (int)(threadIdx.x & 31);
  const int j0 = wave * 16;
  if (j0 >= M) return;  // uniform per-wave exit (after the barrier)

  const int m  = lane & 15;
  const int hi = lane >> 4;
  const int j  = j0 + m;
  const bool jr = (j < M);
  const int lanehn = hi * COUT + m;  // per-lane LDS sub-offset (n chunk added below)

  v8f acc[NCH] = {};

  for (int k = 0; k < 27; ++k) {
    // hit iff scatter index == j (scatter is conflict-free by construction).
    // Index maps are streamed read-once -> non-temporal loads keep L2 for X/W.
    int  gi    = 0;
    bool found = false;
    if (jr) {
      gi    = __builtin_nontemporal_load(G + (long)k * M + j);
      found = (__builtin_nontemporal_load(S + (long)k * M + j) == j);
    }
    const float* xrow = X + (long)gi * CIN;
    const v2f*   wk   = wlds + k * (KSTEPS * 2 * COUT) + lanehn;

#pragma unroll
    for (int kk = 0; kk < KSTEPS; ++kk) {
      // A operand: masked gather of 2 consecutive features (zero row on miss)
      v2f A;
      if (found) {
        A.x = xrow[kk * 4 + 2 * hi + 0];
        A.y = xrow[kk * 4 + 2 * hi + 1];
      } else {
        A.x = 0.0f;
        A.y = 0.0f;
      }
#pragma unroll
      for (int c = 0; c < NCH; ++c) {
        const int n = m + c * 16;
        v2f B;
        if (n < COUT) {  // compile-time true for COUT=32; masks cols for COUT=2
          B = wk[kk * (2 * COUT) + c * 16];  // ds_load_b64, immediate offset
        } else {
          B.x = 0.0f;
          B.y = 0.0f;
        }
        acc[c] = __builtin_amdgcn_wmma_f32_16x16x4_f32(
            false, A, false, B, (short)0, acc[c], false, false);
      }
    }
  }

  // ---- store D tiles (coalesced across lanes within each chunk)
#pragma unroll
  for (int c = 0; c < NCH; ++c) {
    const int n = m + c * 16;
    if (n < COUT) {
#pragma unroll
      for (int r = 0; r < 8; ++r) {
        const int row = j0 + r + 8 * hi;
        if (row < M) Y[(long)row * COUT + n] = acc[c][r];
      }
    }
  }
}

static inline int blocks_for(int M) {
  const int waves = (M + 15) / 16;  // one wave per 16-row tile
  return (waves + 7) / 8;           // 8 waves (256 threads) per block
}

template <int CIN, int COUT>
static constexpr size_t lds_bytes() {
  return (size_t)27 * (CIN / 4) * 2 * COUT * sizeof(v2f);
}

extern "C" void kernel_launch(void* const* d_in, const int* in_sizes, int n_in,
                              void* d_out, int out_size, void* d_ws, size_t ws_size,
                              hipStream_t stream) {
  (void)n_in; (void)out_size; (void)ws_size;
  const float* feats = (const float*)d_in[0];  // [N,4]
  const float* W1    = (const float*)d_in[1];  // [27,4,32]
  const float* W2    = (const float*)d_in[2];  // [27,32,32]
  const float* W3    = (const float*)d_in[3];  // [27,32,2]
  const int*   g1    = (const int*)d_in[4];    // [27,N]
  const int*   s1    = (const int*)d_in[5];
  const int*   g2    = (const int*)d_in[6];    // [27,M2]
  const int*   s2    = (const int*)d_in[7];
  const int*   g3    = (const int*)d_in[8];    // [27,M2]
  const int*   s3    = (const int*)d_in[9];

  const int N  = in_sizes[0] / 4;
  const int M2 = in_sizes[6] / 27;

  float* x1  = (float*)d_ws;                   // [N,32]  (25.6 MB)
  float* x2  = x1 + (size_t)N * 32;            // [M2,32] (~12 MB)
  float* out = (float*)d_out;                  // [M2,2]

  sparse_conv_wmma<4, 32>
      <<<blocks_for(N), 256, lds_bytes<4, 32>(), stream>>>(feats, W1, g1, s1, x1, N);
  sparse_conv_wmma<32, 32>
      <<<blocks_for(M2), 256, lds_bytes<32, 32>(), stream>>>(x1, W2, g2, s2, x2, M2);
  sparse_conv_wmma<32, 2>
      <<<blocks_for(M2), 256, lds_bytes<32, 2>(), stream>>>(x2, W3, g3, s3, out, M2);
}